// SelectiveLossOutput_86011015069830
// MI455X (gfx1250) — compile-verified
//
#include <hip/hip_runtime.h>
#include <hip/hip_bf16.h>

// ---------------------------------------------------------------------------
// Problem constants (from reference)
// ---------------------------------------------------------------------------
constexpr int Bn    = 2048;   // batch
constexpr int Tn    = 200;    // targets per row
constexpr int INn   = 1024;
constexpr int H1n   = 2048;
constexpr int H2n   = 1024;
constexpr int EMBED = 1025;   // H2 + bias column

typedef __attribute__((ext_vector_type(2))) float v2f;
typedef __attribute__((ext_vector_type(8))) float v8f;

// ---------------------------------------------------------------------------
// Tiled f32 WMMA GEMM:  C = relu(A[MxK] * B[KxN] + bias[N])
// Block: 256 threads (8 waves), 128x128 tile. Wave -> 32x64 (2x4 16x16 tiles).
// Software-pipelined: next K-tile is fetched into registers while the current
// K-tile is consumed from LDS by the WMMA pipe.
// ---------------------------------------------------------------------------
#define TILE_M 128
#define TILE_N 128
#define TILE_K 16
#define LDA_PAD 20   // floats; 80-byte rows => 16B-aligned b128 stores, even stride for v2f

__global__ __launch_bounds__(256)
void gemm_bias_relu_wmma(const float* __restrict__ A, const float* __restrict__ Bm,
                         const float* __restrict__ bias, float* __restrict__ C,
                         int M, int K, int N) {
  __shared__ float As[TILE_M * LDA_PAD];   // 128 x 20 floats = 10.0 KB
  __shared__ float Bs[TILE_K * TILE_N];    // 16 x 128 floats =  8.0 KB

  const int tid  = threadIdx.x;
  const int lane = tid & 31;
  const int wid  = tid >> 5;     // 0..7
  const int wm   = wid >> 1;     // 0..3 : 32-row strip within block tile
  const int wn   = wid & 1;      // 0..1 : 64-col strip within block tile
  const int half = lane >> 4;    // lane half selects K pair / M+8
  const int l16  = lane & 15;

  const int m0 = blockIdx.y * TILE_M;
  const int n0 = blockIdx.x * TILE_N;

  v8f acc[2][4];
  #pragma unroll
  for (int i = 0; i < 2; ++i)
    #pragma unroll
    for (int j = 0; j < 4; ++j)
      #pragma unroll
      for (int e = 0; e < 8; ++e) acc[i][j][e] = 0.0f;

  // ---- register-staged global fetch (double buffer vs LDS) ----
  float4 aR[2], bR[2];
  auto stage_load = [&](int k0) {
    #pragma unroll
    for (int p = 0; p < 2; ++p) {
      int idx = tid + p * 256;                // 0..511
      int ar  = idx >> 2, ac = (idx & 3) << 2;    // A: 128 rows x 4 float4
      aR[p] = *(const float4*)(A + (size_t)(m0 + ar) * K + k0 + ac);
      int br  = idx >> 5, bc = (idx & 31) << 2;   // B: 16 rows x 32 float4
      bR[p] = *(const float4*)(Bm + (size_t)(k0 + br) * N + n0 + bc);
    }
  };
  auto stage_store = [&]() {
    #pragma unroll
    for (int p = 0; p < 2; ++p) {
      int idx = tid + p * 256;
      int ar  = idx >> 2, ac = (idx & 3) << 2;
      *(float4*)(&As[ar * LDA_PAD + ac]) = aR[p];
      int br  = idx >> 5, bc = (idx & 31) << 2;
      *(float4*)(&Bs[br * TILE_N + bc]) = bR[p];
    }
  };

  stage_load(0);
  for (int k0 = 0; k0 < K; k0 += TILE_K) {
    stage_store();
    __syncthreads();
    if (k0 + TILE_K < K) stage_load(k0 + TILE_K);   // overlaps with WMMA below

    #pragma unroll
    for (int kk = 0; kk < TILE_K; kk += 4) {
      // A fragment (16x4 f32): lane<16 -> K=kk,kk+1 ; lane>=16 -> K=kk+2,kk+3
      v2f afrag[2];
      #pragma unroll
      for (int i = 0; i < 2; ++i) {
        int row = wm * 32 + i * 16 + l16;
        afrag[i] = *(const v2f*)(&As[row * LDA_PAD + kk + half * 2]);
      }
      // B fragment (4x16 f32): V0 holds rows kk / kk+2 across lane halves
      v2f bfrag[4];
      #pragma unroll
      for (int j = 0; j < 4; ++j) {
        int col = wn * 64 + j * 16 + l16;
        bfrag[j].x = Bs[(kk + half * 2 + 0) * TILE_N + col];
        bfrag[j].y = Bs[(kk + half * 2 + 1) * TILE_N + col];
      }
      #pragma unroll
      for (int i = 0; i < 2; ++i)
        #pragma unroll
        for (int j = 0; j < 4; ++j)
          acc[i][j] = __builtin_amdgcn_wmma_f32_16x16x4_f32(
              false, afrag[i], false, bfrag[j], (short)0, acc[i][j],
              false, false);
    }
    __syncthreads();
  }

  // ---- epilogue: bias + ReLU + store (C/D layout: VGPR r -> M = r + half*8) ----
  #pragma unroll
  for (int j = 0; j < 4; ++j) {
    int col = n0 + wn * 64 + j * 16 + l16;
    float bv = bias[col];
    #pragma unroll
    for (int i = 0; i < 2; ++i) {
      #pragma unroll
      for (int r = 0; r < 8; ++r) {
        int row = m0 + wm * 32 + i * 16 + half * 8 + r;
        float v = acc[i][j][r] + bv;
        C[(size_t)row * N + col] = v > 0.0f ? v : 0.0f;
      }
    }
  }
}

// ---------------------------------------------------------------------------
// Gather + batched dot + sigmoid + weighted BCE partial reduction.
// One block per batch row b; h[b] cached in LDS; each wave owns targets
// t = wid, wid+8, ... ; 32 lanes stride the 1024-dim dot (coalesced b32).
// Next target's embedding row is prefetched (global_prefetch_b8) one
// iteration ahead to overlap the random-gather latency.
// ---------------------------------------------------------------------------
__global__ __launch_bounds__(256)
void gather_dot_loss(const float* __restrict__ h, const float* __restrict__ emb,
                     const int* __restrict__ ids, const float* __restrict__ vals,
                     const float* __restrict__ mask, float* __restrict__ out) {
  __shared__ float hs[H2n];
  __shared__ float wsum[8];

  const int b    = blockIdx.x;
  const int tid  = threadIdx.x;
  const int lane = tid & 31;
  const int wid  = tid >> 5;

  #pragma unroll
  for (int p = 0; p < H2n / 256; ++p)
    hs[tid + p * 256] = h[(size_t)b * H2n + tid + p * 256];
  __syncthreads();

  // prefetch the first row this wave will read
  {
    int id0 = ids[b * Tn + wid];
    if (id0 != 0)
      __builtin_prefetch((const char*)(emb + (size_t)id0 * EMBED) + lane * 128, 0, 0);
  }

  float wacc = 0.0f;  // valid on lane 0 of each wave
  for (int t = wid; t < Tn; t += 8) {
    const int id = ids[b * Tn + t];

    // prefetch next target's row (4100 B ~= 32 lanes x 128 B)
    const int tn = t + 8;
    if (tn < Tn) {
      int idn = ids[b * Tn + tn];
      if (idn != 0)
        __builtin_prefetch((const char*)(emb + (size_t)idn * EMBED) + lane * 128, 0, 0);
    }

    float z = 0.0f;
    if (id != 0) {  // padding_idx row is zero
      const float* te = emb + (size_t)id * EMBED;
      float s = 0.0f;
      #pragma unroll
      for (int c = 0; c < H2n / 32; ++c) {
        int e = c * 32 + lane;        // consecutive lanes -> consecutive floats
        s += te[e] * hs[e];
      }
      #pragma unroll
      for (int off = 16; off > 0; off >>= 1) s += __shfl_down(s, off, 32);
      z = s + te[H2n];                // appended ones-column == bias term
    }
    if (lane == 0) {
      float y = vals[b * Tn + t];
      float m = mask[b * Tn + t];
      float sig = 1.0f / (1.0f + expf(-z));
      float bce = fmaxf(z, 0.0f) - z * y + log1pf(expf(-fabsf(z)));
      out[1 + b * Tn + t] = sig;
      wacc += m * bce;
    }
  }
  if (lane == 0) wsum[wid] = wacc;
  __syncthreads();
  if (tid == 0) {
    float s = 0.0f;
    #pragma unroll
    for (int i = 0; i < 8; ++i) s += wsum[i];
    atomicAdd(out, s * (1.0f / ((float)Bn * (float)Tn)));
  }
}

__global__ void zero_loss(float* out) { out[0] = 0.0f; }

// ---------------------------------------------------------------------------
// Launch
// ---------------------------------------------------------------------------
extern "C" void kernel_launch(void* const* d_in, const int* in_sizes, int n_in,
                              void* d_out, int out_size, void* d_ws, size_t ws_size,
                              hipStream_t stream) {
  const float* x   = (const float*)d_in[0];
  const float* W1  = (const float*)d_in[1];
  const float* b1  = (const float*)d_in[2];
  const float* W2  = (const float*)d_in[3];
  const float* b2  = (const float*)d_in[4];
  const float* emb = (const float*)d_in[5];
  const int*   tid = (const int*)d_in[6];
  const float* tv  = (const float*)d_in[7];
  const float* tm  = (const float*)d_in[8];
  float* out = (float*)d_out;

  float* h1 = (float*)d_ws;                       // [2048 x 2048] = 16 MB
  float* h2 = h1 + (size_t)Bn * H1n;              // [2048 x 1024] =  8 MB

  zero_loss<<<1, 1, 0, stream>>>(out);

  // h1 = relu(x @ W1 + b1)   M=2048 K=1024 N=2048
  gemm_bias_relu_wmma<<<dim3(H1n / TILE_N, Bn / TILE_M), 256, 0, stream>>>(
      x, W1, b1, h1, Bn, INn, H1n);

  // h2 = relu(h1 @ W2 + b2)  M=2048 K=2048 N=1024
  gemm_bias_relu_wmma<<<dim3(H2n / TILE_N, Bn / TILE_M), 256, 0, stream>>>(
      h1, W2, b2, h2, Bn, H1n, H2n);

  // logits / sigmoid / loss
  gather_dot_loss<<<Bn, 256, 0, stream>>>(h2, emb, tid, tv, tm, out);
}